// SimpleGCN_47081431499005
// MI455X (gfx1250) — compile-verified
//
#include <hip/hip_runtime.h>
#include <hip/hip_bf16.h>

typedef __attribute__((ext_vector_type(2))) float v2f;
typedef __attribute__((ext_vector_type(8))) float v8f;

// ---------------------------------------------------------------------------
// native f32 global atomic add (global_atomic_add_f32 on gfx1250)
// ---------------------------------------------------------------------------
__device__ __forceinline__ void atomAddF(float* p, float v) {
  unsafeAtomicAdd(p, v);
}

// ---------------------------------------------------------------------------
// zero a range of floats (float4 granularity)
// ---------------------------------------------------------------------------
__global__ void k_zero(float4* __restrict__ p, long long n4) {
  long long i = (long long)blockIdx.x * blockDim.x + threadIdx.x;
  if (i < n4) p[i] = make_float4(0.f, 0.f, 0.f, 0.f);
}

// ---------------------------------------------------------------------------
// degree accumulation: deg[dst[e]] += 1
// ---------------------------------------------------------------------------
__global__ void k_degree(const int* __restrict__ dst, int E, float* __restrict__ deg) {
  int i = blockIdx.x * blockDim.x + threadIdx.x;
  if (i < E) atomAddF(&deg[dst[i]], 1.0f);
}

// dinv[i] = (deg[i] + 1)^-0.5   (self loop adds 1); in-place over deg buffer
__global__ void k_dinv(float* __restrict__ deg, int N) {
  int i = blockIdx.x * blockDim.x + threadIdx.x;
  if (i < N) deg[i] = rsqrtf(deg[i] + 1.0f);
}

// ---------------------------------------------------------------------------
// out[N,64] = A[N,64] @ W[64,64]  via V_WMMA_F32_16X16X4_F32
// One wave computes a 16x64 output stripe (4 accumulator tiles).
// A fragment: lane&15 = M row; lanes>=16 hold K+2,K+3.  B mirrored (lane&15=N).
// C/D: vgpr r -> M=r (+8 for lanes>=16), N = lane&15.
// ---------------------------------------------------------------------------
__global__ void k_gemm_wmma(const float* __restrict__ A,
                            const float* __restrict__ W,   // row-major [k][n]
                            float* __restrict__ out,
                            int nRowBlocks) {
  const int wave = threadIdx.x >> 5;
  const int lane = threadIdx.x & 31;
  const int rb   = blockIdx.x * (blockDim.x >> 5) + wave;
  if (rb >= nRowBlocks) return;              // wave-uniform: EXEC stays all-1

  const int m     = lane & 15;
  const int khalf = (lane >> 4) << 1;        // 0 or 2

  const float* __restrict__ arow = A + (long long)(rb * 16 + m) * 64;

  v8f acc0 = {}, acc1 = {}, acc2 = {}, acc3 = {};

  #pragma unroll
  for (int k = 0; k < 64; k += 4) {
    v2f a;
    a.x = arow[k + khalf];
    a.y = arow[k + khalf + 1];

    const float* __restrict__ w0 = W + (k + khalf) * 64;
    const float* __restrict__ w1 = w0 + 64;

    v2f b0, b1, b2, b3;
    b0.x = w0[m];      b0.y = w1[m];
    b1.x = w0[16 + m]; b1.y = w1[16 + m];
    b2.x = w0[32 + m]; b2.y = w1[32 + m];
    b3.x = w0[48 + m]; b3.y = w1[48 + m];

    acc0 = __builtin_amdgcn_wmma_f32_16x16x4_f32(false, a, false, b0, (short)0, acc0, false, false);
    acc1 = __builtin_amdgcn_wmma_f32_16x16x4_f32(false, a, false, b1, (short)0, acc1, false, false);
    acc2 = __builtin_amdgcn_wmma_f32_16x16x4_f32(false, a, false, b2, (short)0, acc2, false, false);
    acc3 = __builtin_amdgcn_wmma_f32_16x16x4_f32(false, a, false, b3, (short)0, acc3, false, false);
  }

  const int n     = lane & 15;
  const int mbase = rb * 16 + ((lane >> 4) ? 8 : 0);
  float* __restrict__ orow = out + (long long)mbase * 64;
  #pragma unroll
  for (int r = 0; r < 8; ++r) {
    orow[(long long)r * 64 +  0 + n] = acc0[r];
    orow[(long long)r * 64 + 16 + n] = acc1[r];
    orow[(long long)r * 64 + 32 + n] = acc2[r];
    orow[(long long)r * 64 + 48 + n] = acc3[r];
  }
}

// ---------------------------------------------------------------------------
// edge scatter: hacc[dst] += xw[src] * dinv[src]*dinv[dst]
// one wave per edge; lane covers features {lane, lane+32}
// ---------------------------------------------------------------------------
__global__ void k_prop(const float* __restrict__ xw, const float* __restrict__ dinv,
                       const int* __restrict__ src, const int* __restrict__ dst,
                       float* __restrict__ hacc, int E) {
  long long gid = (long long)blockIdx.x * blockDim.x + threadIdx.x;
  int e    = (int)(gid >> 5);
  int lane = (int)(gid & 31);
  if (e >= E) return;
  int s = src[e], d = dst[e];
  float norm = dinv[s] * dinv[d];
  const float* __restrict__ xs = xw + (long long)s * 64;
  float* __restrict__ hd = hacc + (long long)d * 64;
  atomAddF(&hd[lane],      xs[lane]      * norm);
  atomAddF(&hd[lane + 32], xs[lane + 32] * norm);
}

// ---------------------------------------------------------------------------
// epilogue: h = relu(hacc + xw*dinv^2 (self loop) + bias)   (in-place on hacc)
// ---------------------------------------------------------------------------
__global__ void k_post(float* __restrict__ hacc, const float* __restrict__ xw,
                       const float* __restrict__ dinv, const float* __restrict__ b,
                       long long total) {
  long long i = (long long)blockIdx.x * blockDim.x + threadIdx.x;
  if (i >= total) return;
  int n = (int)(i >> 6), f = (int)(i & 63);
  float di = dinv[n];
  float v  = hacc[i] + xw[i] * di * di + b[f];
  hacc[i]  = fmaxf(v, 0.0f);
}

// ---------------------------------------------------------------------------
// feature-wise sum over nodes: gsum[f] += sum_n h[n*64+f]
// stride is a multiple of 64, so feature for thread t is invariant: t & 63
// ---------------------------------------------------------------------------
__global__ void k_mean(const float* __restrict__ h, float* __restrict__ gsum,
                       long long total) {
  __shared__ float sm[256];
  int t = threadIdx.x;
  float acc = 0.f;
  for (long long i = (long long)blockIdx.x * 256 + t; i < total;
       i += (long long)gridDim.x * 256)
    acc += h[i];
  sm[t] = acc;
  __syncthreads();
  if (t < 64) atomAddF(&gsum[t], sm[t] + sm[t + 64] + sm[t + 128] + sm[t + 192]);
}

// ---------------------------------------------------------------------------
// readout: out[j] = (gsum/N) @ Wr[:,j] + br[j],  j in [0,128)
// ---------------------------------------------------------------------------
__global__ void k_final(const float* __restrict__ gsum, const float* __restrict__ Wr,
                        const float* __restrict__ br, float* __restrict__ out,
                        float invN) {
  int j = threadIdx.x;  // 128 threads
  float acc = 0.f;
  #pragma unroll
  for (int k = 0; k < 64; ++k) acc += (gsum[k] * invN) * Wr[k * 128 + j];
  out[j] = acc + br[j];
}

// ---------------------------------------------------------------------------
// launch
// ---------------------------------------------------------------------------
extern "C" void kernel_launch(void* const* d_in, const int* in_sizes, int n_in,
                              void* d_out, int out_size, void* d_ws, size_t ws_size,
                              hipStream_t stream) {
  const float* x  = (const float*)d_in[0];
  const int*   ei = (const int*)d_in[1];
  const float* W1 = (const float*)d_in[2];
  const float* b1 = (const float*)d_in[3];
  const float* W2 = (const float*)d_in[4];
  const float* b2 = (const float*)d_in[5];
  const float* Wr = (const float*)d_in[6];
  const float* br = (const float*)d_in[7];
  float* out = (float*)d_out;

  const int N = in_sizes[0] / 64;      // 50000
  const int E = in_sizes[1] / 2;       // 800000
  const int* src = ei;
  const int* dst = ei + E;

  const long long NH = (long long)N * 64;
  float* ws = (float*)d_ws;

  // workspace layout (floats), 64-aligned
  const long long offDinv = 0;
  const long long offXw   = ((long long)(N + 63) / 64) * 64;   // shared xw buffer (both layers)
  const long long offH1   = offXw + NH;
  const long long offH2   = offH1 + NH;
  const long long offG    = offH2 + NH;
  const long long totalF  = offG + 64;

  float* dinv = ws + offDinv;          // deg, then dinv in place
  float* xw   = ws + offXw;
  float* h1   = ws + offH1;
  float* h2   = ws + offH2;
  float* gsum = ws + offG;

  const int BS = 256;

  // 0) zero deg/dinv + h1 + h2 + gsum (and xw region, harmless)
  long long n4 = totalF / 4;
  k_zero<<<(unsigned)((n4 + BS - 1) / BS), BS, 0, stream>>>((float4*)ws, n4);

  // 1) degrees + dinv
  k_degree<<<(E + BS - 1) / BS, BS, 0, stream>>>(dst, E, dinv);
  k_dinv<<<(N + BS - 1) / BS, BS, 0, stream>>>(dinv, N);

  const int nRB       = N / 16;                       // 3125 row blocks
  const int wavesPerB = BS / 32;
  const int gemmGrid  = (nRB + wavesPerB - 1) / wavesPerB;
  const long long propT = (long long)E * 32;
  const unsigned propGrid = (unsigned)((propT + BS - 1) / BS);
  const unsigned postGrid = (unsigned)((NH + BS - 1) / BS);

  // 2) layer 1: xw = x@W1 ; scatter ; bias+relu (+self loop)
  k_gemm_wmma<<<gemmGrid, BS, 0, stream>>>(x, W1, xw, nRB);
  k_prop<<<propGrid, BS, 0, stream>>>(xw, dinv, src, dst, h1, E);
  k_post<<<postGrid, BS, 0, stream>>>(h1, xw, dinv, b1, NH);

  // 3) layer 2: xw = h1@W2 ; scatter ; bias+relu (+self loop)
  k_gemm_wmma<<<gemmGrid, BS, 0, stream>>>(h1, W2, xw, nRB);
  k_prop<<<propGrid, BS, 0, stream>>>(xw, dinv, src, dst, h2, E);
  k_post<<<postGrid, BS, 0, stream>>>(h2, xw, dinv, b2, NH);

  // 4) mean pool + readout
  k_mean<<<128, 256, 0, stream>>>(h2, gsum, NH);
  k_final<<<1, 128, 0, stream>>>(gsum, Wr, br, out, 1.0f / (float)N);
}